// GraphEstimator_58342835749134
// MI455X (gfx1250) — compile-verified
//
#include <hip/hip_runtime.h>

typedef __attribute__((ext_vector_type(2))) float v2f;
typedef __attribute__((ext_vector_type(8))) float v8f;

#define F_TOKENS 1024
#define D_TOKEN  64
#define HIDDEN   128

// ---------------------------------------------------------------------------
// Kernel 1: a = node @ W1[:64], bb = node @ W1[64:] + b1, via
// V_WMMA_F32_16X16X4_F32 (full f32 precision). One wave per 16(i) x 16(k)
// output tile; K=64 reduced in 16 chained k=4 WMMA steps. Both accumulators
// share the same A operand (node tile).
//
// Layout assumptions (ISA 7.12.2, 32-bit 16x4 A): lane L -> m = L%16,
// half = L/16; A.v0 = node[m, d0+2*half], A.v1 = node[m, d0+2*half+1].
// B (4x16) assumed dual: lane L -> n = L%16; B.v0 = W1[d0+2*half, n],
// B.v1 = W1[d0+2*half+1, n]. C/D: VGPR r holds M = r + 8*half, N = L%16.
// ---------------------------------------------------------------------------
__global__ __launch_bounds__(32)
void proj_wmma_f32(const float* __restrict__ node, const float* __restrict__ W1,
                   const float* __restrict__ b1, float* __restrict__ Aout,
                   float* __restrict__ Bout) {
  const int i0   = blockIdx.x * 16;   // token tile
  const int k0   = blockIdx.y * 16;   // hidden tile
  const int lane = threadIdx.x;
  const int m    = lane & 15;
  const int half = lane >> 4;

  v8f accA = {};
  v8f accB = {};

#pragma unroll
  for (int dd = 0; dd < D_TOKEN / 4; ++dd) {
    const int d0 = dd * 4;
    const int dr = d0 + 2 * half;

    v2f a;
    a.x = node[(i0 + m) * D_TOKEN + dr];
    a.y = node[(i0 + m) * D_TOKEN + dr + 1];

    v2f bA, bB;
    bA.x = W1[dr * HIDDEN + k0 + m];
    bA.y = W1[(dr + 1) * HIDDEN + k0 + m];
    bB.x = W1[(D_TOKEN + dr) * HIDDEN + k0 + m];
    bB.y = W1[(D_TOKEN + dr + 1) * HIDDEN + k0 + m];

    accA = __builtin_amdgcn_wmma_f32_16x16x4_f32(
        false, a, false, bA, (short)0, accA, false, false);
    accB = __builtin_amdgcn_wmma_f32_16x16x4_f32(
        false, a, false, bB, (short)0, accB, false, false);
  }

  const int   col  = k0 + m;
  const float bias = b1[col];
#pragma unroll
  for (int r = 0; r < 8; ++r) {
    const int row = i0 + r + 8 * half;
    Aout[row * HIDDEN + col] = accA[r];
    Bout[row * HIDDEN + col] = accB[r] + bias;   // fold b1 into b-rows
  }
}

// ---------------------------------------------------------------------------
// Kernel 2: one workgroup (8 wave32) per output row i.
// logits[i,j] = sum_k gelu(a_i[k] + bb_j[k]) * w2[k] + b2, then fused softmax
// over j. a_i and w2 staged in LDS (broadcast reuse); bb rows are L2-resident
// (512 KB total). Output writes coalesced (j = tid + 256*t).
// ---------------------------------------------------------------------------
__global__ __launch_bounds__(256)
void pair_gelu_softmax(const float* __restrict__ Arows,
                       const float* __restrict__ Brows,
                       const float* __restrict__ w2,
                       const float* __restrict__ b2,
                       float* __restrict__ out) {
  __shared__ float a_s[HIDDEN];
  __shared__ float w_s[HIDDEN];
  __shared__ float red[8];

  const int i    = blockIdx.x;
  const int tid  = threadIdx.x;
  const int lane = tid & 31;
  const int wid  = tid >> 5;

  if (tid < HIDDEN)            a_s[tid]          = Arows[i * HIDDEN + tid];
  else if (tid < 2 * HIDDEN)   w_s[tid - HIDDEN] = w2[tid - HIDDEN];
  __syncthreads();

  const float  bias2 = b2[0];
  const float4* a4   = reinterpret_cast<const float4*>(a_s);
  const float4* w4   = reinterpret_cast<const float4*>(w_s);
  const float  kRS2  = 0.70710678118654752f;   // 1/sqrt(2)

  float lg[4];
#pragma unroll
  for (int t = 0; t < 4; ++t) {
    const int     j    = tid + 256 * t;
    const float4* brow = reinterpret_cast<const float4*>(Brows + j * HIDDEN);
    float acc = 0.f;
#pragma unroll 4
    for (int kk = 0; kk < HIDDEN / 4; ++kk) {
      const float4 b = brow[kk];
      const float4 a = a4[kk];
      const float4 w = w4[kk];
      const float x0 = a.x + b.x, x1 = a.y + b.y;
      const float x2 = a.z + b.z, x3 = a.w + b.w;
      acc += 0.5f * x0 * (1.f + erff(x0 * kRS2)) * w.x;   // exact gelu
      acc += 0.5f * x1 * (1.f + erff(x1 * kRS2)) * w.y;
      acc += 0.5f * x2 * (1.f + erff(x2 * kRS2)) * w.z;
      acc += 0.5f * x3 * (1.f + erff(x3 * kRS2)) * w.w;
    }
    lg[t] = acc + bias2;
  }

  // ---- row max (wave32 shuffle + 8-wave LDS reduce) ----
  float mx = fmaxf(fmaxf(lg[0], lg[1]), fmaxf(lg[2], lg[3]));
#pragma unroll
  for (int off = 16; off >= 1; off >>= 1)
    mx = fmaxf(mx, __shfl_xor(mx, off, 32));
  if (lane == 0) red[wid] = mx;
  __syncthreads();
#pragma unroll
  for (int w = 0; w < 8; ++w) mx = fmaxf(mx, red[w]);

  // ---- exp + row sum ----
  float e[4];
  float s = 0.f;
#pragma unroll
  for (int t = 0; t < 4; ++t) { e[t] = expf(lg[t] - mx); s += e[t]; }
#pragma unroll
  for (int off = 16; off >= 1; off >>= 1)
    s += __shfl_xor(s, off, 32);
  __syncthreads();               // everyone done reading red (max) before reuse
  if (lane == 0) red[wid] = s;
  __syncthreads();
  float tot = 0.f;
#pragma unroll
  for (int w = 0; w < 8; ++w) tot += red[w];
  const float inv = 1.f / tot;

#pragma unroll
  for (int t = 0; t < 4; ++t)
    out[i * F_TOKENS + tid + 256 * t] = e[t] * inv;   // coalesced
}

extern "C" void kernel_launch(void* const* d_in, const int* in_sizes, int n_in,
                              void* d_out, int out_size, void* d_ws, size_t ws_size,
                              hipStream_t stream) {
  const float* node = (const float*)d_in[0];   // [1024, 64]
  const float* W1   = (const float*)d_in[1];   // [128, 128]
  const float* b1   = (const float*)d_in[2];   // [128]
  const float* w2   = (const float*)d_in[3];   // [128]
  const float* b2   = (const float*)d_in[4];   // [1]
  float*       out  = (float*)d_out;           // [1024, 1024]

  float* Aout = (float*)d_ws;                      // [1024, 128]
  float* Bout = Aout + (size_t)F_TOKENS * HIDDEN;  // [1024, 128] (b1 folded)

  proj_wmma_f32<<<dim3(F_TOKENS / 16, HIDDEN / 16), 32, 0, stream>>>(
      node, W1, b1, Aout, Bout);
  pair_gelu_softmax<<<dim3(F_TOKENS), 256, 0, stream>>>(
      Aout, Bout, w2, b2, out);
}